// GAT_6116033429577
// MI455X (gfx1250) — compile-verified
//
#include <hip/hip_runtime.h>
#include <math.h>

// ---------- CDNA5 WMMA types ----------
typedef __attribute__((ext_vector_type(2))) float v2f;
typedef __attribute__((ext_vector_type(8))) float v8f;

#define TPB 256
static constexpr float kNegSlope = 0.2f;

__device__ __forceinline__ float leakyf(float x) { return x > 0.f ? x : kNegSlope * x; }
__device__ __forceinline__ float eluf(float x)   { return x > 0.f ? x : expm1f(x); }

// float atomic max via signed/unsigned integer ordering trick
__device__ __forceinline__ void atomicMaxF(float* addr, float v) {
  if (v >= 0.f) atomicMax((int*)addr, __float_as_int(v));
  else          atomicMin((unsigned int*)addr, __float_as_uint(v));
}

// ---------- small utility kernels ----------
__global__ void k_zero(float* __restrict__ p, int n) {
  int i = blockIdx.x * blockDim.x + threadIdx.x;
  if (i < n) p[i] = 0.f;
}

// zero-pad [N,7] feature matrix into [N,8] so the WMMA GEMM needs no K guards
__global__ void k_pad_x(const float* __restrict__ x, float* __restrict__ x8, int N) {
  int i = blockIdx.x * blockDim.x + threadIdx.x;
  if (i >= N * 8) return;
  int r = i >> 3, c = i & 7;
  x8[i] = (c < 7) ? x[r * 7 + c] : 0.f;
}

// sum of edge attributes -> scal[0]
__global__ void k_edge_stats(const float* __restrict__ ea, int E, float* scal) {
  __shared__ float s[TPB];
  float acc = 0.f;
  for (int i = blockIdx.x * blockDim.x + threadIdx.x; i < E; i += gridDim.x * blockDim.x)
    acc += ea[i];
  s[threadIdx.x] = acc;
  __syncthreads();
  for (int w = TPB / 2; w > 0; w >>= 1) {
    if ((int)threadIdx.x < w) s[threadIdx.x] += s[threadIdx.x + w];
    __syncthreads();
  }
  if (threadIdx.x == 0) atomicAdd(&scal[0], s[0]);
}

// scal[1] = dot(We, ae); scal[2] = mean(eattr)*ce  (self-loop edge logit)
__global__ void k_ce(const float* __restrict__ We, const float* __restrict__ ae,
                     int dout, int E, float* scal) {
  if (blockIdx.x == 0 && threadIdx.x == 0) {
    float ce = 0.f;
    for (int j = 0; j < dout; ++j) ce += We[j] * ae[j];
    scal[1] = ce;
    scal[2] = (scal[0] / (float)E) * ce;
  }
}

// ---------- WMMA fp32 GEMM: H[M,NC] = X[M,K] @ W[K,NC] ----------
// One block per 16-row strip; NC/16 waves each own one 16x16 output tile.
// A tile (16 x K) staged in LDS cooperatively (coalesced), read back as
// aligned ds_load_b64 fragments. K is compile-time -> fully unrolled chain
// of V_WMMA_F32_16X16X4_F32 with unguarded loads (EXEC all-ones throughout).
template <int K>
__global__ __launch_bounds__(TPB)
void k_gemm_wmma(const float* __restrict__ X, int ldx,
                 const float* __restrict__ W, int ldw,
                 float* __restrict__ H, int ldh) {
  __shared__ float sA[16 * K];
  const int tm = blockIdx.x;

  // cooperative, coalesced load of the 16 x K A-tile
  for (int idx = threadIdx.x; idx < 16 * K; idx += blockDim.x) {
    int r = idx / K, c = idx - r * K;
    sA[idx] = X[(size_t)(tm * 16 + r) * ldx + c];
  }
  __syncthreads();

  const int lane = threadIdx.x & 31;
  const int wave = threadIdx.x >> 5;     // column tile index, 0..NC/16-1
  const int l16  = lane & 15;
  const int kg   = lane >> 4;            // half-wave selects K pair
  const int n    = wave * 16 + l16;      // B/D column this lane supplies

  v8f c = {};
#pragma unroll
  for (int k0 = 0; k0 < K; k0 += 4) {
    const int ka = k0 + 2 * kg;
    v2f a = *(const v2f*)&sA[l16 * K + ka];      // aligned: K and ka both even
    v2f b;
    b.x = W[(size_t)ka * ldw + n];
    b.y = W[(size_t)(ka + 1) * ldw + n];
    // (neg_a, A, neg_b, B, c_mod, C, reuse_a, reuse_b)
    c = __builtin_amdgcn_wmma_f32_16x16x4_f32(false, a, false, b, (short)0, c,
                                              false, false);
  }

  // D layout: VGPR r holds row tm*16 + r + 8*kg, column n
  const int mb = tm * 16 + kg * 8;
#pragma unroll
  for (int r = 0; r < 8; ++r)
    H[(size_t)(mb + r) * ldh + n] = c[r];
}

// ---------- attention passes ----------
// per-node attention logit pieces: als = h.a_s, ald = h.a_d
__global__ void k_node_alpha(const float* __restrict__ H, int ldh,
                             const float* __restrict__ a_s, const float* __restrict__ a_d,
                             float* __restrict__ als, float* __restrict__ ald,
                             int N, int dout) {
  int n = blockIdx.x * blockDim.x + threadIdx.x;
  if (n >= N) return;
  const float* h = H + (size_t)n * ldh;
  float s = 0.f, d = 0.f;
  for (int j = 0; j < dout; ++j) { float v = h[j]; s += v * a_s[j]; d += v * a_d[j]; }
  als[n] = s; ald[n] = d;
}

// mx[n] = self-loop alpha (every node has a self-loop)
__global__ void k_node_mx_init(const float* __restrict__ als, const float* __restrict__ ald,
                               const float* __restrict__ scal, float* __restrict__ mx, int N) {
  int n = blockIdx.x * blockDim.x + threadIdx.x;
  if (n >= N) return;
  mx[n] = leakyf(als[n] + ald[n] + scal[2]);
}

__global__ void k_edge_max(const int* __restrict__ src, const int* __restrict__ dst,
                           const float* __restrict__ ea,
                           const float* __restrict__ als, const float* __restrict__ ald,
                           const float* __restrict__ scal, float* __restrict__ mx, int E) {
  int e = blockIdx.x * blockDim.x + threadIdx.x;
  if (e >= E) return;
  float a = leakyf(als[src[e]] + ald[dst[e]] + ea[e] * scal[1]);
  atomicMaxF(&mx[dst[e]], a);
}

// exs[n] = exp(alpha_self - mx); den[n] = exs[n]; zero the output accumulator row
__global__ void k_node_prep(const float* __restrict__ als, const float* __restrict__ ald,
                            const float* __restrict__ scal, const float* __restrict__ mx,
                            float* __restrict__ exs, float* __restrict__ den,
                            float* __restrict__ O, int ldo, int dout, int N) {
  int n = blockIdx.x * blockDim.x + threadIdx.x;
  if (n >= N) return;
  float aself = leakyf(als[n] + ald[n] + scal[2]);
  float ex = expf(aself - mx[n]);
  exs[n] = ex;
  den[n] = ex;
  float* o = O + (size_t)n * ldo;
  for (int j = 0; j < dout; ++j) o[j] = 0.f;
}

__global__ void k_edge_exp(const int* __restrict__ src, const int* __restrict__ dst,
                           const float* __restrict__ ea,
                           const float* __restrict__ als, const float* __restrict__ ald,
                           const float* __restrict__ scal, const float* __restrict__ mx,
                           float* __restrict__ exe, float* __restrict__ den, int E) {
  int e = blockIdx.x * blockDim.x + threadIdx.x;
  if (e >= E) return;
  int d = dst[e];
  float a = leakyf(als[src[e]] + ald[d] + ea[e] * scal[1]);
  float ex = expf(a - mx[d]);
  exe[e] = ex;
  atomicAdd(&den[d], ex);
}

// one block per edge (blockDim == dout); scatter coef * h[src] into O[dst]
__global__ void k_edge_scatter(const int* __restrict__ src, const int* __restrict__ dst,
                               const float* __restrict__ exe, const float* __restrict__ den,
                               const float* __restrict__ H, int ldh,
                               float* __restrict__ O, int ldo, int E) {
  int e = blockIdx.x;
  if (e >= E) return;
  int s = src[e], d = dst[e];
  float coef = exe[e] / den[d];
  int j = threadIdx.x;
  atomicAdd(&O[(size_t)d * ldo + j], coef * H[(size_t)s * ldh + j]);
}

// add self-loop term + bias, apply ELU in place
__global__ void k_node_final(const float* __restrict__ exs, const float* __restrict__ den,
                             const float* __restrict__ H, int ldh,
                             const float* __restrict__ b,
                             float* __restrict__ O, int ldo, int dout, int N) {
  int i = blockIdx.x * blockDim.x + threadIdx.x;
  if (i >= N * dout) return;
  int n = i / dout, j = i - n * dout;
  float coef = exs[n] / den[n];
  float v = O[(size_t)n * ldo + j] + coef * H[(size_t)n * ldh + j] + b[j];
  O[(size_t)n * ldo + j] = eluf(v);
}

// column-sum pool over first dcols columns (dcols=64)
__global__ void k_pool(const float* __restrict__ X, int ld, int dcols, int N,
                       float* __restrict__ pool) {
  __shared__ float s[TPB];
  int j = threadIdx.x % dcols;
  int r = threadIdx.x / dcols;
  int rowsPerBlk = blockDim.x / dcols;
  float acc = 0.f;
  for (int n = blockIdx.x * rowsPerBlk + r; n < N; n += gridDim.x * rowsPerBlk)
    acc += X[(size_t)n * ld + j];
  s[threadIdx.x] = acc;
  __syncthreads();
  if (r == 0) {
    for (int q = 1; q < rowsPerBlk; ++q) acc += s[q * dcols + j];
    atomicAdd(&pool[j], acc);
  }
}

// out[j] = b_lin[j] + sum_i<64 (pool[i]/N) Wl[i,j] + sum_i<16 xn[i] Wl[64+i,j]
__global__ void k_final_linear(const float* __restrict__ pool, float invN,
                               const float* __restrict__ xn,
                               const float* __restrict__ Wl, const float* __restrict__ bl,
                               float* __restrict__ out) {
  int j = threadIdx.x;
  if (j >= 64) return;
  float v = bl[j];
  for (int i = 0; i < 64; ++i) v += (pool[i] * invN) * Wl[i * 64 + j];
  for (int i = 0; i < 16; ++i) v += xn[i] * Wl[(64 + i) * 64 + j];
  out[j] = v;
}

// ---------- host orchestration ----------
extern "C" void kernel_launch(void* const* d_in, const int* in_sizes, int n_in,
                              void* d_out, int out_size, void* d_ws, size_t ws_size,
                              hipStream_t stream) {
  const float* x[2]  = { (const float*)d_in[0], (const float*)d_in[1] };
  const int*   ei[2] = { (const int*)d_in[2],   (const int*)d_in[3]   };
  const float* xn[2] = { (const float*)d_in[4], (const float*)d_in[5] };
  const float* ec[2] = { (const float*)d_in[6], (const float*)d_in[7] };

  const int N = in_sizes[0] / 7;   // 50000 (= 3125 * 16, tile-exact)
  const int E = in_sizes[2] / 2;   // 600000

  const float* Wm[3]  = { (const float*)d_in[8],  (const float*)d_in[14], (const float*)d_in[20] };
  const float* asv[3] = { (const float*)d_in[9],  (const float*)d_in[15], (const float*)d_in[21] };
  const float* adv[3] = { (const float*)d_in[10], (const float*)d_in[16], (const float*)d_in[22] };
  const float* Wev[3] = { (const float*)d_in[11], (const float*)d_in[17], (const float*)d_in[23] };
  const float* aev[3] = { (const float*)d_in[12], (const float*)d_in[18], (const float*)d_in[24] };
  const float* bv[3]  = { (const float*)d_in[13], (const float*)d_in[19], (const float*)d_in[25] };
  const float* Wlin = (const float*)d_in[26];
  const float* blin = (const float*)d_in[27];
  const int douts[3] = { 128, 128, 64 };

  // workspace carve (floats)
  float* p     = (float*)d_ws;
  float* bufA  = p; p += (size_t)N * 128;   // H  (x @ W)
  float* bufB  = p; p += (size_t)N * 128;   // attention out / next-layer input
  float* bufX8 = p; p += (size_t)N * 8;     // zero-padded layer-1 input
  float* als   = p; p += N;
  float* ald   = p; p += N;
  float* mx    = p; p += N;
  float* den   = p; p += N;
  float* exs   = p; p += N;
  float* exe   = p; p += E;
  float* pool  = p; p += 64;
  float* scal  = p; p += 8;                 // [0]=sum(eattr) [1]=ce [2]=al_e_self

  const int nb_N = (N + TPB - 1) / TPB;
  const int nb_E = (E + TPB - 1) / TPB;

  for (int g = 0; g < 2; ++g) {
    const int* src = ei[g];
    const int* dst = ei[g] + E;

    // reset pool + scalars each call (graph-replay safe)
    k_zero<<<1, 128, 0, stream>>>(pool, 64 + 8);
    k_edge_stats<<<256, TPB, 0, stream>>>(ec[g], E, scal);
    k_pad_x<<<(N * 8 + TPB - 1) / TPB, TPB, 0, stream>>>(x[g], bufX8, N);

    const float* Xcur = bufX8;
    for (int L = 0; L < 3; ++L) {
      const int dout = douts[L];

      k_ce<<<1, 32, 0, stream>>>(Wev[L], aev[L], dout, E, scal);

      // H = Xcur @ W  via fp32 WMMA: one block per 16-row strip,
      // dout/16 waves per block, compile-time K (fully unrolled WMMA chain)
      if (L == 0)
        k_gemm_wmma<8><<<N / 16, (dout / 16) * 32, 0, stream>>>(Xcur, 8, Wm[L], dout,
                                                                bufA, 128);
      else
        k_gemm_wmma<128><<<N / 16, (dout / 16) * 32, 0, stream>>>(Xcur, 128, Wm[L], dout,
                                                                  bufA, 128);

      k_node_alpha<<<nb_N, TPB, 0, stream>>>(bufA, 128, asv[L], adv[L], als, ald, N, dout);
      k_node_mx_init<<<nb_N, TPB, 0, stream>>>(als, ald, scal, mx, N);
      k_edge_max<<<nb_E, TPB, 0, stream>>>(src, dst, ec[g], als, ald, scal, mx, E);
      k_node_prep<<<nb_N, TPB, 0, stream>>>(als, ald, scal, mx, exs, den, bufB, 128, dout, N);
      k_edge_exp<<<nb_E, TPB, 0, stream>>>(src, dst, ec[g], als, ald, scal, mx, exe, den, E);
      k_edge_scatter<<<E, dout, 0, stream>>>(src, dst, exe, den, bufA, 128, bufB, 128, E);

      const int tot = N * dout;
      k_node_final<<<(tot + TPB - 1) / TPB, TPB, 0, stream>>>(exs, den, bufA, 128, bv[L],
                                                              bufB, 128, dout, N);
      Xcur = bufB;
    }

    // graph mean-pool (64 cols) + final linear -> d_out[g*64 .. g*64+63]
    k_pool<<<256, TPB, 0, stream>>>(bufB, 128, 64, N, pool);
    k_final_linear<<<1, 64, 0, stream>>>(pool, 1.f / (float)N, xn[g], Wlin, blin,
                                         (float*)d_out + g * 64);
  }
}